// FeedForwardNet_Time_83648783057347
// MI455X (gfx1250) — compile-verified
//
#include <hip/hip_runtime.h>

#define NB 8
#define NC 256
#define NL 4096
#define NK 8
#define BIGF 1e20f
#define PAD 8
#define NT 512                 // threads per block (16 waves of 32)
#define CG 4                   // channels per workgroup (softmax weights shared)
#define NG 2                   // groups of 4 consecutive l per thread (NT*4*NG == NL)
#define NW (NT / 32)           // waves per block

#if defined(__AMDGCN__) && __has_builtin(__builtin_amdgcn_global_load_async_to_lds_b128)
#define HAVE_ASYNC_LDS 1
#else
#define HAVE_ASYNC_LDS 0
#endif

#if defined(__AMDGCN__) && __has_builtin(__builtin_amdgcn_s_wait_asynccnt)
#define WAIT_ASYNC() __builtin_amdgcn_s_wait_asynccnt(0)
#else
#define WAIT_ASYNC() asm volatile("s_wait_asynccnt 0" ::: "memory")
#endif

// Native clang vector types (HIP's float4 is a class; builtins need these).
typedef float v4f __attribute__((ext_vector_type(4)));
typedef int v4i __attribute__((ext_vector_type(4)));
typedef __attribute__((address_space(1))) v4i gv4i;
typedef __attribute__((address_space(3))) v4i lv4i;

// One workgroup per (b, 4-channel group). Stages dt row + 4 x rows into LDS
// via async global->LDS, computes the K=8 softmax weights ONCE per l and
// applies them to all 4 channels. Windowed LDS reads are 3 aligned b128 loads
// per 4 consecutive l. Block-reduces per-channel 2x2 Gram, blends, stores b128.
// __launch_bounds__(NT, 1): allow full VGPR budget -> no scratch spills.
__global__ __launch_bounds__(NT, 1) void fused_time_attn(
    const float* __restrict__ x,      // (B, C, L)
    const float* __restrict__ dt,     // (B, L)
    const float* __restrict__ ker,    // (1, 1, K)
    float* __restrict__ out)          // (B, C, L)
{
  const int t = threadIdx.x;
  const int cbase = blockIdx.x * CG;
  const int b = blockIdx.y;

  __shared__ __attribute__((aligned(16))) float xs[CG][PAD + NL];  // zero-padded front
  __shared__ __attribute__((aligned(16))) float ts[PAD + NL];      // BIG-padded front
  __shared__ float red[CG][3][NW];
  __shared__ float coef[CG][2];

  const size_t rowbase = ((size_t)b * NC + cbase) * NL;
  const float* __restrict__ drow = dt + (size_t)b * NL;

  if (t == 0) {
    __builtin_prefetch(drow, 0, 3);           // global_prefetch_b8
#pragma unroll
    for (int c = 0; c < CG; ++c) __builtin_prefetch(x + rowbase + (size_t)c * NL, 0, 3);
  }
  if (t < PAD) {
    ts[t] = BIGF;                             // rolled dt is BIG where masked
#pragma unroll
    for (int c = 0; c < CG; ++c) xs[c][t] = 0.0f;  // rolled x is 0 where masked
  }

#pragma unroll
  for (int j = 0; j < NL / (4 * NT); ++j) {   // 2 float4's per thread per row
    const int v4 = j * NT + t;
#if HAVE_ASYNC_LDS
    __builtin_amdgcn_global_load_async_to_lds_b128(
        (gv4i*)(drow + v4 * 4), (lv4i*)(&ts[PAD + v4 * 4]), 0, 0);
#pragma unroll
    for (int c = 0; c < CG; ++c)
      __builtin_amdgcn_global_load_async_to_lds_b128(
          (gv4i*)(x + rowbase + (size_t)c * NL + v4 * 4),
          (lv4i*)(&xs[c][PAD + v4 * 4]), 0, 0);
#else
    *(v4f*)(&ts[PAD + v4 * 4]) = ((const v4f*)drow)[v4];
#pragma unroll
    for (int c = 0; c < CG; ++c)
      *(v4f*)(&xs[c][PAD + v4 * 4]) = ((const v4f*)(x + rowbase + (size_t)c * NL))[v4];
#endif
  }
#if HAVE_ASYNC_LDS
  WAIT_ASYNC();
#endif
  __syncthreads();

  float kreg[NK];
#pragma unroll
  for (int k = 0; k < NK; ++k) kreg[k] = ker[k];  // uniform -> scalar loads

  float a0r[CG][4 * NG], a1r[CG][4 * NG];
  float g00[CG], g01[CG], g11[CG];
#pragma unroll
  for (int c = 0; c < CG; ++c) { g00[c] = 0.f; g01[c] = 0.f; g11[c] = 0.f; }

#pragma unroll
  for (int g = 0; g < NG; ++g) {
    // This thread covers l = base+j, j=0..3. Window ts/xs[base .. base+11]
    // (word index base+m == PAD + (base-8) + m), 16B aligned since base%4==0.
    const int base = g * (4 * NT) + 4 * t;

    float Wd[12];
#pragma unroll
    for (int q = 0; q < 3; ++q)
      *(v4f*)&Wd[4 * q] = *(const v4f*)&ts[base + 4 * q];

    // Softmax weights per j (shift-invariant: dt[l] cancels against the max).
    float e[4][NK], inv_se[4];
#pragma unroll
    for (int j = 0; j < 4; ++j) {
      float m = Wd[j + 1];
#pragma unroll
      for (int q = 2; q <= 8; ++q) m = fmaxf(m, Wd[j + q]);
      float se = 0.f;
#pragma unroll
      for (int k = 0; k < NK; ++k) {
        // dv_k = Wd[8+j-k]; masked: exp(BIG-BIG)=1, others underflow to 0.
        const float ev = __expf(Wd[j + 8 - k] - m);
        e[j][k] = ev;
        se += ev;
      }
      inv_se[j] = 1.0f / se;
    }

#pragma unroll
    for (int c = 0; c < CG; ++c) {
      float Wx[12];
#pragma unroll
      for (int q = 0; q < 3; ++q)
        *(v4f*)&Wx[4 * q] = *(const v4f*)&xs[c][base + 4 * q];
#pragma unroll
      for (int j = 0; j < 4; ++j) {
        float a0 = 0.f, a1 = 0.f;
#pragma unroll
        for (int k = 0; k < NK; ++k) {
          const float xv = Wx[j + 8 - k];   // x[l-k] (0 where masked)
          a0 = fmaf(xv, e[j][k], a0);
          a1 = fmaf(xv, kreg[k], a1);
        }
        a0 *= inv_se[j];
        a0r[c][g * 4 + j] = a0;
        a1r[c][g * 4 + j] = a1;
        g00[c] = fmaf(a0, a0, g00[c]);
        g01[c] = fmaf(a0, a1, g01[c]);
        g11[c] = fmaf(a1, a1, g11[c]);
      }
    }
  }

  // ---- block reduction of per-channel Gram entries (wave32 shuffle + LDS) ----
  const int lane = t & 31;
  const int wid = t >> 5;
#pragma unroll
  for (int c = 0; c < CG; ++c) {
#pragma unroll
    for (int off = 16; off >= 1; off >>= 1) {
      g00[c] += __shfl_xor(g00[c], off, 32);
      g01[c] += __shfl_xor(g01[c], off, 32);
      g11[c] += __shfl_xor(g11[c], off, 32);
    }
    if (lane == 0) {
      red[c][0][wid] = g00[c];
      red[c][1][wid] = g01[c];
      red[c][2][wid] = g11[c];
    }
  }
  __syncthreads();

  if (t < CG) {  // one thread finalizes each channel's 2x2 softmax
    float d00 = 0.f, d01 = 0.f, d11 = 0.f;
#pragma unroll
    for (int w = 0; w < NW; ++w) {
      d00 += red[t][0][w];
      d01 += red[t][1][w];
      d11 += red[t][2][w];
    }
    const float m0 = fmaxf(d00, d01);
    const float e00 = __expf(d00 - m0), e01 = __expf(d01 - m0);
    const float r0 = 1.0f / (e00 + e01);
    const float m1 = fmaxf(d01, d11);
    const float e10 = __expf(d01 - m1), e11 = __expf(d11 - m1);
    const float r1 = 1.0f / (e10 + e11);
    coef[t][0] = 0.5f * (e00 * r0 + e10 * r1);
    coef[t][1] = 0.5f * (e01 * r0 + e11 * r1);
  }
  __syncthreads();

  // ---- blend and store (b128, coalesced, non-temporal: out is never re-read) ----
#pragma unroll
  for (int c = 0; c < CG; ++c) {
    const float c0 = coef[c][0];
    const float c1 = coef[c][1];
    float* __restrict__ orow = out + rowbase + (size_t)c * NL;
#pragma unroll
    for (int g = 0; g < NG; ++g) {
      const int base = g * (4 * NT) + 4 * t;
      v4f v;
      v.x = fmaf(c0, a0r[c][g * 4 + 0], c1 * a1r[c][g * 4 + 0]);
      v.y = fmaf(c0, a0r[c][g * 4 + 1], c1 * a1r[c][g * 4 + 1]);
      v.z = fmaf(c0, a0r[c][g * 4 + 2], c1 * a1r[c][g * 4 + 2]);
      v.w = fmaf(c0, a0r[c][g * 4 + 3], c1 * a1r[c][g * 4 + 3]);
      __builtin_nontemporal_store(v, (v4f*)&orow[base]);
    }
  }
}

extern "C" void kernel_launch(void* const* d_in, const int* in_sizes, int n_in,
                              void* d_out, int out_size, void* d_ws, size_t ws_size,
                              hipStream_t stream) {
  (void)in_sizes; (void)n_in; (void)out_size; (void)d_ws; (void)ws_size;
  const float* x   = (const float*)d_in[0];   // (8, 256, 4096) f32
  const float* dt  = (const float*)d_in[1];   // (8, 4096) f32
  const float* ker = (const float*)d_in[2];   // (1, 1, 8) f32
  float* out = (float*)d_out;                 // (8, 256, 4096) f32

  dim3 grid(NC / CG, NB);  // one workgroup per (b, 4 channels)
  fused_time_attn<<<grid, NT, 0, stream>>>(x, dt, ker, out);
}